// Involution_23794118820103
// MI455X (gfx1250) — compile-verified
//
#include <hip/hip_runtime.h>

typedef float v2f __attribute__((ext_vector_type(2)));
typedef float v8f __attribute__((ext_vector_type(8)));

#define C_    256      // channels
#define CR    64       // C / reduction_ratio
#define G_    16       // groups
#define KK    49       // 7*7
#define KKG   784      // 49*16
#define NPIX  16384    // 4*64*64
#define BN_EPS 1e-3f

// ---------------------------------------------------------------------------
// GEMM1: t = relu(BN(x @ w1 + b1))   (16384 x 256) @ (256 x 64) -> (16384 x 64)
// One wave32 computes one 16x16 tile via V_WMMA_F32_16X16X4_F32.
// A layout (16x4 f32): lane l(0..15)/l+16 holds M=l, K=2*hi+{0,1} in 2 VGPRs.
// B layout (4x16 f32): mirrored. C/D: VGPR r -> M=r+8*hi, N=l.
// ---------------------------------------------------------------------------
__global__ __launch_bounds__(256) void inv_gemm1(
    const float* __restrict__ x,  const float* __restrict__ w1,
    const float* __restrict__ b1, const float* __restrict__ gamma,
    const float* __restrict__ beta, const float* __restrict__ mean,
    const float* __restrict__ var, float* __restrict__ t)
{
    const int wave = threadIdx.x >> 5;
    const int lane = threadIdx.x & 31;
    const int tile = blockIdx.x * 8 + wave;
    const int m0 = (tile >> 2) * 16;          // 4 N-tiles (64/16)
    const int n0 = (tile & 3) * 16;
    const int l  = lane & 15;
    const int hi = lane >> 4;

    v8f acc = {};
    const float* arow = x  + (size_t)(m0 + l) * C_ + 2 * hi;   // A: row m0+l
    const float* bcol = w1 + (size_t)(2 * hi) * CR + (n0 + l); // B: col n0+l
    #pragma unroll 4
    for (int k0 = 0; k0 < C_; k0 += 4) {
        v2f a = *(const v2f*)(arow + k0);      // K = k0+2*hi, k0+2*hi+1
        v2f b;
        b.x = bcol[(size_t)k0 * CR];
        b.y = bcol[(size_t)k0 * CR + CR];
        acc = __builtin_amdgcn_wmma_f32_16x16x4_f32(
            false, a, false, b, (short)0, acc, false, false);
    }

    // Fold BN (inference) + bias + ReLU into the epilogue.
    const int col = n0 + l;
    const float s  = gamma[col] * rsqrtf(var[col] + BN_EPS);
    const float sh = (b1[col] - mean[col]) * s + beta[col];
    #pragma unroll
    for (int r = 0; r < 8; ++r) {
        const int row = m0 + r + 8 * hi;
        t[(size_t)row * CR + col] = fmaxf(acc[r] * s + sh, 0.0f);
    }
}

// ---------------------------------------------------------------------------
// GEMM2: kernel = t @ w2 + b2   (16384 x 64) @ (64 x 784) -> (16384 x 784)
// N = 784 = 49 exact 16-wide tiles. 16 WMMAs per tile.
// ---------------------------------------------------------------------------
__global__ __launch_bounds__(256) void inv_gemm2(
    const float* __restrict__ t, const float* __restrict__ w2,
    const float* __restrict__ b2, float* __restrict__ kout)
{
    const int wave = threadIdx.x >> 5;
    const int lane = threadIdx.x & 31;
    const int tile = blockIdx.x * 8 + wave;
    const int m0 = (tile / 49) * 16;
    const int n0 = (tile % 49) * 16;
    const int l  = lane & 15;
    const int hi = lane >> 4;

    v8f acc = {};
    const float* arow = t  + (size_t)(m0 + l) * CR + 2 * hi;
    const float* bcol = w2 + (size_t)(2 * hi) * KKG + (n0 + l);
    #pragma unroll
    for (int k0 = 0; k0 < CR; k0 += 4) {
        v2f a = *(const v2f*)(arow + k0);
        v2f b;
        b.x = bcol[(size_t)k0 * KKG];
        b.y = bcol[(size_t)k0 * KKG + KKG];
        acc = __builtin_amdgcn_wmma_f32_16x16x4_f32(
            false, a, false, b, (short)0, acc, false, false);
    }

    const int col  = n0 + l;
    const float bb = b2[col];
    #pragma unroll
    for (int r = 0; r < 8; ++r) {
        const int row = m0 + r + 8 * hi;
        kout[(size_t)row * KKG + col] = acc[r] + bb;
    }
}

// ---------------------------------------------------------------------------
// Involution gather: one block per pixel, thread = channel c (g = c % 16).
// out[pix][c] = sum_k ker[pix][k*16+g] * x_pad[b][h+ki-3][w+kj-3][c]
// 784 per-pixel kernel weights staged in LDS (shared across 16 chans/group).
// ---------------------------------------------------------------------------
__global__ __launch_bounds__(256) void inv_apply(
    const float* __restrict__ x, const float* __restrict__ ker,
    float* __restrict__ out)
{
    __shared__ float sk[KKG];
    const int pix = blockIdx.x;
    const int c   = threadIdx.x;

    const float* krow = ker + (size_t)pix * KKG;
    for (int i = threadIdx.x; i < KKG; i += 256) sk[i] = krow[i];
    __syncthreads();

    const int b = pix >> 12;
    const int h = (pix >> 6) & 63;
    const int w = pix & 63;
    const int g = c & (G_ - 1);
    const float* xb = x + ((size_t)b << 12) * C_;

    float acc = 0.0f;
    #pragma unroll
    for (int ki = 0; ki < 7; ++ki) {
        const int hh = h + ki - 3;
        if ((unsigned)hh >= 64u) continue;
        #pragma unroll
        for (int kj = 0; kj < 7; ++kj) {
            const int ww = w + kj - 3;
            if ((unsigned)ww >= 64u) continue;
            const float kv = sk[(ki * 7 + kj) * G_ + g];
            const float xv = xb[(size_t)((hh << 6) + ww) * C_ + c];
            acc = fmaf(kv, xv, acc);
        }
    }
    out[(size_t)pix * C_ + c] = acc;
}

// ---------------------------------------------------------------------------
extern "C" void kernel_launch(void* const* d_in, const int* in_sizes, int n_in,
                              void* d_out, int out_size, void* d_ws, size_t ws_size,
                              hipStream_t stream)
{
    const float* x     = (const float*)d_in[0];
    const float* w1    = (const float*)d_in[1];
    const float* b1    = (const float*)d_in[2];
    const float* gamma = (const float*)d_in[3];
    const float* beta  = (const float*)d_in[4];
    const float* mean  = (const float*)d_in[5];
    const float* var   = (const float*)d_in[6];
    const float* w2    = (const float*)d_in[7];
    const float* b2    = (const float*)d_in[8];

    float* out  = (float*)d_out;                     // (B,H,W,256)
    float* kout = out + (size_t)NPIX * C_;           // (B,H,W,784) second output
    float* t    = (float*)d_ws;                      // (16384 x 64) fp32 = 4 MB

    // GEMM1: 1024*4 tiles, 8 waves/block -> 512 blocks
    inv_gemm1<<<512, 256, 0, stream>>>(x, w1, b1, gamma, beta, mean, var, t);
    // GEMM2: 1024*49 tiles, 8 waves/block -> 6272 blocks
    inv_gemm2<<<6272, 256, 0, stream>>>(t, w2, b2, kout);
    // Involution: one block per pixel
    inv_apply<<<NPIX, 256, 0, stream>>>(x, kout, out);
}